// EfficentLePE_9466107920974
// MI455X (gfx1250) — compile-verified
//
#include <hip/hip_runtime.h>

typedef float v2f __attribute__((ext_vector_type(2)));
typedef float v4f __attribute__((ext_vector_type(4)));
typedef float v8f __attribute__((ext_vector_type(8)));

#define BB    8
#define RES   128
#define DIMC  128
#define NHEAD 8
#define HD    16
#define WSP   8
#define LTOK  (RES*RES)
#define SCALE 0.25f

// One workgroup per (window, head). 256 threads = 8 waves (wave32).
__global__ __launch_bounds__(256) void cswin_lepe_kernel(
    const float* __restrict__ qkv,
    const float* __restrict__ conv_w,
    const float* __restrict__ conv_b,
    float* __restrict__ out)
{
    const int w    = blockIdx.x;      // 0..127 window
    const int nh   = blockIdx.y;      // 0..7 head
    const int b    = w >> 4;          // batch
    const int wblk = w & 15;          // horizontal window block
    const int tid  = threadIdx.x;
    const int lane = tid & 31;
    const int wave = tid >> 5;        // 0..7

    const float* qp = qkv;
    const float* kp = qkv + (size_t)BB * LTOK * DIMC;
    const float* vp = qkv + (size_t)2 * BB * LTOK * DIMC;

    // (n, d) of this window/head lives at base + ((n>>3)*RES + (n&7))*DIMC + d
    const size_t base = (size_t)b * LTOK * DIMC + (size_t)wblk * WSP * DIMC
                      + (size_t)nh * HD;

    __shared__ float s_red[8 * 256];     // per-wave partial 16x16 accumulators
    __shared__ float s_att[16 * 16];     // softmaxed attention matrix
    __shared__ float s_q[8][4 * 8 * 16]; // per-wave q staging: 4 h-rows x 8 wsub x 16 ch

    const int col  = lane & 15;         // matrix column / channel within head
    const int rsel = (lane >> 4) * 2;   // K-slot select per A/B f32 layout (0 or 2)

    // ---------------- Phase A: att_raw = K^T V (contract over N=1024) ----------------
    v8f acc = {0.f, 0.f, 0.f, 0.f, 0.f, 0.f, 0.f, 0.f};
    for (int i = 0; i < 32; ++i) {
        const int n0 = wave * 128 + i * 4;     // K-chunk base token
        const int ra = n0 + rsel;
        const int rb = ra + 1;
        const size_t off0 = base + (size_t)(((ra >> 3) * RES) + (ra & 7)) * DIMC + col;
        const size_t off1 = base + (size_t)(((rb >> 3) * RES) + (rb & 7)) * DIMC + col;
        v2f a, bv;
        a.x  = kp[off0];  a.y  = kp[off1];     // A = K^T : A[m,kk] = K[n0+kk, m]
        bv.x = vp[off0];  bv.y = vp[off1];     // B = V   : B[kk,e] = V[n0+kk, e]
        acc = __builtin_amdgcn_wmma_f32_16x16x4_f32(
                  false, a, false, bv, (short)0, acc, false, false);
    }
    #pragma unroll
    for (int r = 0; r < 8; ++r)
        s_red[wave * 256 + r * 32 + lane] = acc[r];
    __syncthreads();

    // reduce across 8 waves, scale, scatter into [M][N] layout
    {
        const int r = tid >> 5;
        const int l = tid & 31;
        float v = 0.f;
        #pragma unroll
        for (int i = 0; i < 8; ++i) v += s_red[i * 256 + r * 32 + l];
        const int M = r + 8 * (l >> 4);
        const int Nc = l & 15;
        s_att[M * 16 + Nc] = v * SCALE;
    }
    __syncthreads();

    // row softmax (axis = e)
    if (tid < 16) {
        float mx = -3.402823e38f;
        #pragma unroll
        for (int e = 0; e < 16; ++e) mx = fmaxf(mx, s_att[tid * 16 + e]);
        float tmp[16];
        float s = 0.f;
        #pragma unroll
        for (int e = 0; e < 16; ++e) {
            const float t = __expf(s_att[tid * 16 + e] - mx);
            tmp[e] = t; s += t;
        }
        const float inv = 1.f / s;
        #pragma unroll
        for (int e = 0; e < 16; ++e) s_att[tid * 16 + e] = tmp[e] * inv;
    }
    __syncthreads();

    // depthwise conv weights for this lane's channel (e = col)
    const int ch = nh * HD + col;
    float w9[9];
    #pragma unroll
    for (int j = 0; j < 9; ++j) w9[j] = conv_w[ch * 9 + j];
    const float bias = conv_b[ch];

    // ---------------- Phase B: x = Q*att + LePE, per-wave 16-row tiles ----------------
    float* sq = s_q[wave];
    const int Mbase = 8 * (lane >> 4);

    for (int t = 0; t < 8; ++t) {
        const int m = t * 8 + wave;            // tile id 0..63 -> rows n = 16m..16m+15
        const int hbase = 2 * m - 1;           // staged h-rows: hbase..hbase+3 (halo)

        // stage 4 h-rows x 8 wsub x 16 channels via async global->LDS copies;
        // image-boundary rows are zero-filled with plain DS stores (disjoint addrs).
        // Async LDS writes are unordered vs DS ops, so drain DScnt first.
        asm volatile("s_wait_dscnt 0x0" ::: "memory");
        for (int c4 = lane; c4 < 128; c4 += 32) {
            const int hr  = c4 >> 5;
            const int rem = c4 & 31;
            const int ws  = rem >> 2;
            const int d4  = (rem & 3) * 4;
            const int h   = hbase + hr;
            if (h >= 0 && h < RES) {
                // low 32 bits of a flat shared pointer = LDS byte offset
                const unsigned ldsa =
                    (unsigned)(uintptr_t)(const void*)(sq + c4 * 4);
                const float* ga = qp + base + (size_t)(h * RES + ws) * DIMC + d4;
                asm volatile("global_load_async_to_lds_b128 %0, %1, off"
                             :: "v"(ldsa), "v"(ga) : "memory");
            } else {
                const v4f z = {0.f, 0.f, 0.f, 0.f};
                *(v4f*)(sq + c4 * 4) = z;
            }
        }
        asm volatile("s_wait_asynccnt 0x0" ::: "memory");
        __syncthreads();

        // D = Qtile(16x16) * att(16x16) via 4 chained f32 WMMAs
        v8f d = {0.f, 0.f, 0.f, 0.f, 0.f, 0.f, 0.f, 0.f};
        {
            const int Mr = lane & 15;               // A row
            const int hr = (Mr >> 3) + 1;           // buffer h-row (skip halo)
            const int ws = Mr & 7;
            #pragma unroll
            for (int kk0 = 0; kk0 < 16; kk0 += 4) {
                const int cidx = kk0 + rsel;
                v2f a = *(const v2f*)(sq + ((hr * 8 + ws) * 16 + cidx));
                v2f bv;
                bv.x = s_att[(cidx + 0) * 16 + col];
                bv.y = s_att[(cidx + 1) * 16 + col];
                d = __builtin_amdgcn_wmma_f32_16x16x4_f32(
                        false, a, false, bv, (short)0, d, false, false);
            }
        }

        // epilogue: add LePE (3x3 depthwise from LDS) and store
        #pragma unroll
        for (int r = 0; r < 8; ++r) {
            const int M  = r + Mbase;
            const int hr = (M >> 3) + 1;
            const int ws = M & 7;
            float conv = bias;
            #pragma unroll
            for (int dh = -1; dh <= 1; ++dh) {
                #pragma unroll
                for (int dw = -1; dw <= 1; ++dw) {
                    const int wsx = ws + dw;
                    float val = 0.f;
                    if (wsx >= 0 && wsx < 8)
                        val = sq[((hr + dh) * 8 + wsx) * 16 + col];
                    conv = fmaf(val, w9[(dh + 1) * 3 + (dw + 1)], conv);
                }
            }
            const int n = m * 16 + M;
            out[base + (size_t)((n >> 3) * RES + (n & 7)) * DIMC + col] = d[r] + conv;
        }
        __syncthreads();
    }
}

extern "C" void kernel_launch(void* const* d_in, const int* in_sizes, int n_in,
                              void* d_out, int out_size, void* d_ws, size_t ws_size,
                              hipStream_t stream) {
    const float* qkv    = (const float*)d_in[0];   // [3, B, L, DIM] fp32
    const float* conv_w = (const float*)d_in[1];   // [DIM, 1, 3, 3] fp32
    const float* conv_b = (const float*)d_in[2];   // [DIM] fp32
    float* outp = (float*)d_out;                   // [B, L, DIM] fp32
    (void)in_sizes; (void)n_in; (void)out_size; (void)d_ws; (void)ws_size;

    dim3 grid(128, 8, 1);   // (window, head)
    dim3 block(256, 1, 1);  // 8 waves
    cswin_lepe_kernel<<<grid, block, 0, stream>>>(qkv, conv_w, conv_b, outp);
}